// Nagy_47321949667529
// MI455X (gfx1250) — compile-verified
//
#include <hip/hip_runtime.h>
#include <math.h>

typedef __attribute__((ext_vector_type(2))) float v2f;
typedef __attribute__((ext_vector_type(8))) float v8f;

#define THREADS 256

// ---------------------------------------------------------------- utilities

__global__ void zero_kernel(float* p, long n) {
  long i = (long)blockIdx.x * blockDim.x + threadIdx.x;
  if (i < n) p[i] = 0.f;
}

template <int COLS>
__global__ void pad_copy_kernel(float* __restrict__ dst, const float* __restrict__ src,
                                int rows, int rows_pad) {
  long i = (long)blockIdx.x * blockDim.x + threadIdx.x;
  if (i >= (long)rows_pad * COLS) return;
  int r = (int)(i / COLS);
  dst[i] = (r < rows) ? src[i] : 0.f;
}

// segment_sum: buf[dst[e]] += x[src[e]]  (buf pre-initialized with the residual)
template <int COLS>
__global__ void scatter_add_kernel(float* __restrict__ buf, const float* __restrict__ x,
                                   const int* __restrict__ esrc, const int* __restrict__ edst,
                                   int nE) {
  long i = (long)blockIdx.x * blockDim.x + threadIdx.x;
  if (i >= (long)nE * COLS) return;
  int e = (int)(i / COLS);
  int c = (int)(i % COLS);
  unsafeAtomicAdd(&buf[(long)edst[e] * COLS + c], x[(long)esrc[e] * COLS + c]);
}

// ---------------------------------------------------------------- f32 WMMA GEMM
// C[i,j] = act( sum_k A[i,k] * W[j,k] + b[j] ),  A:[M,K] padded to 16 rows,
// W:[128,K] row-major, out:[M,128].  One wave per 16x16 tile, 8 waves/block.
template <int K, int ACT>
__global__ __launch_bounds__(256) void gemm_kernel(const float* __restrict__ A,
                                                   const float* __restrict__ W,
                                                   const float* __restrict__ bias,
                                                   float* __restrict__ out, int M) {
  const int lane = threadIdx.x & 31;
  const int wave = threadIdx.x >> 5;    // 0..7 -> N tile
  const int half = lane >> 4;           // 0: K+0/1, 1: K+2/3
  const int lr   = lane & 15;
  const int row0 = blockIdx.x * 16;
  const int n0   = wave * 16;

  const float* a_ptr = A + (long)(row0 + lr) * K + 2 * half;  // A row for this lane
  const float* b_ptr = W + (long)(n0 + lr) * K + 2 * half;    // W row = output column

  v8f acc = {0.f, 0.f, 0.f, 0.f, 0.f, 0.f, 0.f, 0.f};
#pragma unroll 8
  for (int k0 = 0; k0 < K; k0 += 4) {
    v2f a; a.x = a_ptr[k0]; a.y = a_ptr[k0 + 1];
    v2f b; b.x = b_ptr[k0]; b.y = b_ptr[k0 + 1];
    acc = __builtin_amdgcn_wmma_f32_16x16x4_f32(false, a, false, b, (short)0, acc,
                                                false, false);
  }
  const int col = n0 + lr;
  const float bv = bias[col];
#pragma unroll
  for (int r = 0; r < 8; ++r) {
    int rowm = row0 + r + 8 * half;     // D layout: VGPR r -> M=r (lanes 0-15) / 8+r
    float v = acc[r] + bv;
    if (ACT) v = fmaxf(v, 0.f);
    if (rowm < M) out[(long)rowm * 128 + col] = v;
  }
}

// ---------------------------------------------------------------- pipeline kernels

__global__ void concat_init_kernel(float* __restrict__ dst, const float* __restrict__ hq,
                                   const float* __restrict__ hg, int nq, int ntot,
                                   int rows_pad) {
  long i = (long)blockIdx.x * blockDim.x + threadIdx.x;
  if (i >= (long)rows_pad * 128) return;
  int r = (int)(i >> 7), c = (int)(i & 127);
  float v = 0.f;
  if (r < nq) v = hq[i];
  else if (r < ntot) v = hg[(long)(r - nq) * 128 + c];
  dst[i] = v;
}

__global__ __launch_bounds__(256) void gather_out_kernel(float* __restrict__ x1,
                                                         float* __restrict__ x2,
                                                         float* __restrict__ ys,
                                                         const float* __restrict__ itg,
                                                         const int* __restrict__ itedge,
                                                         const int* __restrict__ npair,
                                                         int Eit, int Enp) {
  long j = blockIdx.x;
  if (j >= (long)Eit + Enp) return;
  int t = threadIdx.x;
  if (t == 0) ys[j] = (j < Eit) ? 1.f : -1.f;
  int i1, i2;
  if (j < Eit) { i1 = itedge[j]; i2 = itedge[Eit + j]; }
  else { long k = j - Eit; i1 = npair[k]; i2 = npair[Enp + k]; }
  if (t < 128) x1[j * 128 + t] = itg[(long)i1 * 128 + t];
  else         x2[j * 128 + (t - 128)] = itg[(long)i2 * 128 + (t - 128)];
}

__global__ __launch_bounds__(128) void rowsum_kernel(const float* __restrict__ x,
                                                     float* __restrict__ out,
                                                     int row0, int nrows) {
  int t = threadIdx.x;
  long r0 = (long)blockIdx.x * 128;
  long rend = r0 + 128; if (rend > nrows) rend = nrows;
  float a = 0.f;
  for (long r = r0; r < rend; ++r) a += x[(row0 + r) * 128 + t];
  unsafeAtomicAdd(&out[t], a);
}

__global__ void build_hq2_kernel(float* __restrict__ hq2, const float* __restrict__ itg,
                                 const float* __restrict__ hq1, int nqo, int Nq) {
  int i = blockIdx.x * blockDim.x + threadIdx.x;
  if (i >= Nq * 128) return;
  int r = i >> 7;
  hq2[i] = (r < nqo) ? itg[i] : hq1[i];
}

__global__ void mask_rows_kernel(float* __restrict__ hq2, const int* __restrict__ matched,
                                 int nM) {
  int i = blockIdx.x * blockDim.x + threadIdx.x;
  if (i >= nM * 128) return;
  int r = matched[i >> 7];
  hq2[r * 128 + (i & 127)] = 0.f;
}

__global__ void swish_add_kernel(float* __restrict__ o, const float* __restrict__ a,
                                 const float* __restrict__ b, int n) {
  int i = blockIdx.x * blockDim.x + threadIdx.x;
  if (i >= n) return;
  float z = a[i] + b[i];
  o[i] = z / (1.f + expf(-z));   // swish
}

// out[seg[j]] += W @ X[nodes[j]] + b   (one block per gathered row)
__global__ __launch_bounds__(128) void lin_gather_scatter_kernel(
    const float* __restrict__ Wm, const float* __restrict__ bm,
    const float* __restrict__ X, const int* __restrict__ nodes,
    const int* __restrict__ seg, int nrows, float* __restrict__ out) {
  int j = blockIdx.x;
  if (j >= nrows) return;
  int t = threadIdx.x;
  const float* xr = X + (long)nodes[j] * 128;
  const float* w  = Wm + (long)t * 128;
  float a = bm[t];
  for (int k = 0; k < 128; ++k) a += w[k] * xr[k];
  unsafeAtomicAdd(&out[(long)seg[j] * 128 + t], a);
}

// ---------------------------------------------------------------- finale (single block)
__global__ __launch_bounds__(256) void finale_kernel(
    const float* __restrict__ rq0, const float* __restrict__ sbuf,
    const int* __restrict__ ov_i, const int* __restrict__ ov_j,
    const float* __restrict__ sum_hg,
    const float* __restrict__ W2, const float* __restrict__ b2,
    const float* __restrict__ Wag0, const float* __restrict__ bag0,
    const float* __restrict__ Wag1, const float* __restrict__ bag1,
    const float* __restrict__ Ww, const float* __restrict__ bw,
    const float* __restrict__ Wag2, const float* __restrict__ bag2,
    const float* __restrict__ West, const float* __restrict__ best,
    const float* __restrict__ Ww2, const float* __restrict__ bw2,
    float NgF, float ngoF, float* __restrict__ out1, float* __restrict__ pj_out,
    int ns, int nov) {
  __shared__ float rq0s[8 * 128], ssm[8 * 128], ofm[8 * 128];
  __shared__ float xb[8 * 256], tb[8 * 256];
  __shared__ float hb[8 * 128], t2b[8 * 128], of2b[8 * 128];
  __shared__ float rg[128], rq[128], rf[128];
  __shared__ float cnt[8], wl[8], wts[8];
  __shared__ int ovi_s[8], ovj_s[8];
  int tid = threadIdx.x;

  for (int i = tid; i < ns * 128; i += 256) { rq0s[i] = rq0[i]; ofm[i] = 0.f; }
  for (int i = tid; i < nov * 128; i += 256) ssm[i] = sbuf[i];
  if (tid < nov) { ovi_s[tid] = ov_i[tid]; ovj_s[tid] = ov_j[tid]; }
  if (tid < ns) cnt[tid] = 1.f;
  __syncthreads();

  if (tid < 128) {            // readout_g = W2 @ sum_hg + ng_o * b2
    float a = 0.f;
    for (int k = 0; k < 128; ++k) a += W2[tid * 128 + k] * sum_hg[k];
    rg[tid] = a + ngoF * b2[tid];
  }
  if (tid < 128) {            // overlap scatter (column-owner, serial over overlaps)
    for (int o = 0; o < nov; ++o) {
      float sv = ssm[o * 128 + tid];
      ofm[ovi_s[o] * 128 + tid] += sv;
      ofm[ovj_s[o] * 128 + tid] += sv;
    }
  }
  if (tid == 0)
    for (int o = 0; o < nov; ++o) { cnt[ovi_s[o]] += 1.f; cnt[ovj_s[o]] += 1.f; }
  __syncthreads();

  for (int i = tid; i < ns * 128; i += 256) {   // divide + build x0 = [rq0 | of]
    int r = i >> 7, c = i & 127;
    xb[r * 256 + c] = rq0s[i];
    xb[r * 256 + 128 + c] = ofm[i] / cnt[r];
  }
  __syncthreads();
  for (int i = tid; i < ns * 256; i += 256) tb[i] = xb[i];
  __syncthreads();
  // skeleton edges: flat [i0,j0,...,i7,j7]; src = first nov, dst = last nov
  if (tid < 256) {
    for (int k = 0; k < nov; ++k) {
      int m0 = k, m1 = nov + k;
      int es = (m0 & 1) ? ovj_s[m0 >> 1] : ovi_s[m0 >> 1];
      int ed = (m1 & 1) ? ovj_s[m1 >> 1] : ovi_s[m1 >> 1];
      tb[ed * 256 + tid] += xb[es * 256 + tid];
    }
  }
  __syncthreads();
  for (int i = tid; i < ns * 128; i += 256) {   // aggregate layer 1 (256->128, relu)
    int r = i >> 7, c = i & 127;
    float a = bag0[c];
    const float* w = Wag0 + c * 256;
    const float* x = tb + r * 256;
    for (int k = 0; k < 256; ++k) a += w[k] * x[k];
    hb[i] = fmaxf(a, 0.f);
  }
  __syncthreads();
  for (int i = tid; i < ns * 128; i += 256) t2b[i] = hb[i];
  __syncthreads();
  if (tid < 128) {
    for (int k = 0; k < nov; ++k) {
      int m0 = k, m1 = nov + k;
      int es = (m0 & 1) ? ovj_s[m0 >> 1] : ovi_s[m0 >> 1];
      int ed = (m1 & 1) ? ovj_s[m1 >> 1] : ovi_s[m1 >> 1];
      t2b[ed * 128 + tid] += hb[es * 128 + tid];
    }
  }
  __syncthreads();
  for (int i = tid; i < ns * 128; i += 256) {   // aggregate layer 2 (128->128, relu)
    int r = i >> 7, c = i & 127;
    float a = bag1[c];
    for (int k = 0; k < 128; ++k) a += Wag1[c * 128 + k] * t2b[r * 128 + k];
    of2b[i] = fmaxf(a, 0.f);
  }
  __syncthreads();
  if (tid < ns) {             // weighter logits over hskeleton = [rq0 | of2]
    float a = bw[0];
    for (int k = 0; k < 128; ++k) a += Ww[k] * rq0s[tid * 128 + k];
    for (int k = 0; k < 128; ++k) a += Ww[128 + k] * of2b[tid * 128 + k];
    wl[tid] = a;
  }
  __syncthreads();
  if (tid == 0) {             // softmax over subqueries
    float mx = wl[0];
    for (int r = 1; r < ns; ++r) mx = fmaxf(mx, wl[r]);
    float s = 0.f;
    for (int r = 0; r < ns; ++r) { wts[r] = expf(wl[r] - mx); s += wts[r]; }
    for (int r = 0; r < ns; ++r) wts[r] /= s;
  }
  __syncthreads();
  if (tid < 128) {
    float a = 0.f;
    for (int r = 0; r < ns; ++r) a += rq0s[r * 128 + tid] * wts[r];
    rq[tid] = a;
  }
  __syncthreads();
  if (tid == 0) {             // pjlength
    float n2 = 0.f, dp = 0.f;
    for (int c = 0; c < 128; ++c) { n2 += rq[c] * rq[c]; dp += rq[c] * rg[c]; }
    float nrm = sqrtf(n2);
    pj_out[0] = dp / ((nrm == 0.f) ? 1.f : nrm);
  }
  if (tid < 128) {            // readout_f = aggregate2([rq | rg])
    float a = bag2[tid];
    for (int k = 0; k < 128; ++k) a += Wag2[tid * 256 + k] * rq[k];
    for (int k = 0; k < 128; ++k) a += Wag2[tid * 256 + 128 + k] * rg[k];
    rf[tid] = a;
  }
  __syncthreads();
  if (tid == 0) {             // out1 = estimate(rf) * sigmoid(weighter2([rf, Ng]))
    float e = best[0];
    for (int k = 0; k < 128; ++k) e += West[k] * rf[k];
    float w2 = bw2[0];
    for (int k = 0; k < 128; ++k) w2 += Ww2[k] * rf[k];
    w2 += Ww2[128] * NgF;
    out1[0] = e * (1.f / (1.f + expf(-w2)));
  }
}

// ---------------------------------------------------------------- host launcher

extern "C" void kernel_launch(void* const* d_in, const int* in_sizes, int n_in,
                              void* d_out_v, int out_size, void* d_ws, size_t ws_size,
                              hipStream_t stream) {
  const float* xg = (const float*)d_in[0];
  const int*   eg = (const int*)d_in[1];
  const float* xq = (const float*)d_in[2];
  const int*   eq = (const int*)d_in[3];
  const int*   matched = (const int*)d_in[4];
  const int*   sub_nodes = (const int*)d_in[5];
  const int*   sub_seg = (const int*)d_in[6];
  const int*   ov_i = (const int*)d_in[7];
  const int*   ov_j = (const int*)d_in[8];
  const int*   ov_nodes = (const int*)d_in[9];
  const int*   ov_seg = (const int*)d_in[10];
  const int*   itedge = (const int*)d_in[11];
  const int*   npair = (const int*)d_in[12];

  // params: flattened either in insertion order or jax-sorted-key order.
  // Disambiguate by in_sizes[13]: graph_gnn.W0 is 128*64, aggregate.W0 is 128*256.
  const float *Wg0, *bg0, *Wg1, *bg1, *Wq1, *bq1, *Wq2, *bq2, *Wit0, *bit0, *Wit1, *bit1,
      *Wag0, *bag0, *Wag1, *bag1, *W1, *b1, *W2, *b2, *W3, *b3, *Wag2, *bag2,
      *West, *best, *Ww, *bw, *Ww2, *bw2;
  const float* F[30];
  for (int i = 0; i < 30; ++i) F[i] = (const float*)d_in[13 + i];
  if (in_sizes[13] == 128 * 64) {  // insertion order
    Wg0 = F[0];  bg0 = F[1];  Wg1 = F[2];  bg1 = F[3];
    Wq1 = F[4];  bq1 = F[5];  Wq2 = F[6];  bq2 = F[7];
    Wit0 = F[8]; bit0 = F[9]; Wit1 = F[10]; bit1 = F[11];
    Wag0 = F[12]; bag0 = F[13]; Wag1 = F[14]; bag1 = F[15];
    W1 = F[16]; b1 = F[17]; W2 = F[18]; b2 = F[19]; W3 = F[20]; b3 = F[21];
    Wag2 = F[22]; bag2 = F[23]; West = F[24]; best = F[25];
    Ww = F[26]; bw = F[27]; Ww2 = F[28]; bw2 = F[29];
  } else {                          // alphabetical (jax dict flattening)
    Wag0 = F[0]; bag0 = F[1]; Wag1 = F[2]; bag1 = F[3];
    Wag2 = F[4]; bag2 = F[5]; West = F[6]; best = F[7];
    Wg0 = F[8];  bg0 = F[9];  Wg1 = F[10]; bg1 = F[11];
    Wit0 = F[12]; bit0 = F[13]; Wit1 = F[14]; bit1 = F[15];
    W1 = F[16]; b1 = F[17]; W2 = F[18]; b2 = F[19]; W3 = F[20]; b3 = F[21];
    Wq1 = F[22]; bq1 = F[23]; Wq2 = F[24]; bq2 = F[25];
    Ww = F[26]; bw = F[27]; Ww2 = F[28]; bw2 = F[29];
  }

  const int Ng = in_sizes[0] / 64;   // 55000
  const int Eg = in_sizes[1] / 2;    // 800000
  const int Nq = in_sizes[2] / 64;   // 32
  const int Eq = in_sizes[3] / 2;    // 80
  const int nM = in_sizes[4];        // 12
  const int nSubN = in_sizes[5];     // 48
  const int nOv = in_sizes[7];       // 8
  const int nOvN = in_sizes[9];      // 32
  const int Eit = in_sizes[11] / 2;  // 200000
  const int Enp = in_sizes[12] / 2;  // 200000
  const int ng_o = 50000, nq_o = 24, n_subq = 6;  // fixed scalars from setup_inputs

  const int Ngp = ((Ng + 15) / 16) * 16;
  const int Nit = nq_o + ng_o;
  const int Nitp = ((Nit + 15) / 16) * 16;
  const long S = (long)Ngp * 128;

  float* ws = (float*)d_ws;
  float* slot0 = ws;
  float* slot1 = ws + S;
  float* slot2 = ws + 2 * S;
  float* sm = ws + 3 * S;
  float* tmpq1 = sm + 0;       // Nq*64
  float* hq1   = sm + 2048;    // Nq*128
  float* hq2   = sm + 6144;
  float* tmpq2 = sm + 10240;
  float* g2buf = sm + 14336;
  float* hq3   = sm + 18432;
  float* sum_hg = sm + 22528;  // 128
  float* rq0buf = sm + 22656;  // 8*128
  float* sbuf2  = sm + 23680;  // 8*128
  const long SMALL_TOTAL = 24704;

  float* d_out = (float*)d_out_v;
  const long NE = (long)Eit + Enp;
  float* x1  = d_out + 1;
  float* x2  = x1 + NE * 128;
  float* ysp = x2 + NE * 128;
  float* pjp = d_out + (out_size - 1);

  auto nb = [](long n) { return (unsigned)((n + THREADS - 1) / THREADS); };

  zero_kernel<<<nb(SMALL_TOTAL), THREADS, 0, stream>>>(sm, SMALL_TOTAL);

  // ---- graph GIN layer 1 (64 -> 128)
  pad_copy_kernel<64><<<nb((long)Ngp * 64), THREADS, 0, stream>>>(slot0, xg, Ng, Ngp);
  scatter_add_kernel<64><<<nb((long)Eg * 64), THREADS, 0, stream>>>(slot0, xg, eg, eg + Eg, Eg);
  gemm_kernel<64, 1><<<Ngp / 16, 256, 0, stream>>>(slot0, Wg0, bg0, slot1, Ng);
  // ---- graph GIN layer 2 (128 -> 128)
  pad_copy_kernel<128><<<nb((long)Ngp * 128), THREADS, 0, stream>>>(slot2, slot1, Ng, Ngp);
  scatter_add_kernel<128><<<nb((long)Eg * 128), THREADS, 0, stream>>>(slot2, slot1, eg, eg + Eg, Eg);
  gemm_kernel<128, 1><<<Ngp / 16, 256, 0, stream>>>(slot2, Wg1, bg1, slot0, Ng);  // hg2 -> slot0

  // ---- query GIN 1 (64 -> 128)
  pad_copy_kernel<64><<<nb((long)Nq * 64), THREADS, 0, stream>>>(tmpq1, xq, Nq, Nq);
  scatter_add_kernel<64><<<nb((long)Eq * 64), THREADS, 0, stream>>>(tmpq1, xq, eq, eq + Eq, Eq);
  gemm_kernel<64, 1><<<Nq / 16, 256, 0, stream>>>(tmpq1, Wq1, bq1, hq1, Nq);

  // ---- interactor GIN (2 layers on [hq[:nq_o] ; hg[:ng_o]])
  concat_init_kernel<<<nb((long)Nitp * 128), THREADS, 0, stream>>>(slot1, hq1, slot0, nq_o, Nit, Nitp);
  pad_copy_kernel<128><<<nb((long)Nitp * 128), THREADS, 0, stream>>>(slot2, slot1, Nit, Nitp);
  scatter_add_kernel<128><<<nb((long)Eit * 128), THREADS, 0, stream>>>(slot2, slot1, itedge, itedge + Eit, Eit);
  gemm_kernel<128, 1><<<Nitp / 16, 256, 0, stream>>>(slot2, Wit0, bit0, slot0, Nit);
  pad_copy_kernel<128><<<nb((long)Nitp * 128), THREADS, 0, stream>>>(slot1, slot0, Nit, Nitp);
  scatter_add_kernel<128><<<nb((long)Eit * 128), THREADS, 0, stream>>>(slot1, slot0, itedge, itedge + Eit, Eit);
  gemm_kernel<128, 1><<<Nitp / 16, 256, 0, stream>>>(slot1, Wit1, bit1, slot2, Nit);  // itg -> slot2

  // ---- outputs x1s/x2s/ys (bandwidth-dominant stage)
  gather_out_kernel<<<(unsigned)NE, 256, 0, stream>>>(x1, x2, ysp, slot2, itedge, npair, Eit, Enp);

  // ---- readout_g pre-reduction: sum of itg rows [nq_o, nq_o+ng_o)
  rowsum_kernel<<<(unsigned)((ng_o + 127) / 128), 128, 0, stream>>>(slot2, sum_hg, nq_o, ng_o);

  // ---- query side: hq = [itg[:nq_o] ; hq1[nq_o:]], mask, GIN2, swish residual
  build_hq2_kernel<<<nb((long)Nq * 128), THREADS, 0, stream>>>(hq2, slot2, hq1, nq_o, Nq);
  mask_rows_kernel<<<nb((long)nM * 128), THREADS, 0, stream>>>(hq2, matched, nM);
  pad_copy_kernel<128><<<nb((long)Nq * 128), THREADS, 0, stream>>>(tmpq2, hq2, Nq, Nq);
  scatter_add_kernel<128><<<nb((long)Eq * 128), THREADS, 0, stream>>>(tmpq2, hq2, eq, eq + Eq, Eq);
  gemm_kernel<128, 1><<<Nq / 16, 256, 0, stream>>>(tmpq2, Wq2, bq2, g2buf, Nq);
  swish_add_kernel<<<nb((long)Nq * 128), THREADS, 0, stream>>>(hq3, hq2, g2buf, Nq * 128);

  // ---- per-subquery / overlap segment readouts
  lin_gather_scatter_kernel<<<nSubN, 128, 0, stream>>>(W1, b1, hq3, sub_nodes, sub_seg, nSubN, rq0buf);
  lin_gather_scatter_kernel<<<nOvN, 128, 0, stream>>>(W3, b3, hq3, ov_nodes, ov_seg, nOvN, sbuf2);

  // ---- finale: overlap normalization, skeleton GIN, softmax weighting, scalars
  finale_kernel<<<1, 256, 0, stream>>>(rq0buf, sbuf2, ov_i, ov_j, sum_hg,
                                       W2, b2, Wag0, bag0, Wag1, bag1, Ww, bw,
                                       Wag2, bag2, West, best, Ww2, bw2,
                                       (float)Ng, (float)ng_o, d_out, pjp, n_subq, nOv);
}